// AAAProtectedClassifier_23991687315792
// MI455X (gfx1250) — compile-verified
//
#include <hip/hip_runtime.h>
#include <math.h>

typedef __attribute__((ext_vector_type(2))) float v2f;
typedef __attribute__((ext_vector_type(8))) float v8f;
typedef int v4i_gcc __attribute__((vector_size(4 * sizeof(int))));

#define BB_ 4096
#define DD_ 2048
#define CC_ 1000
#define NTILE 63              /* ceil(1000/16) */
#define KSLAB 32
#define ALPHA_ 1.0f
#define TAU_ 6.0f
#define BETA_ 5.0f
#define LR_ 0.1f
#define WD_ 0.01f
#define AB1_ 0.9f
#define AB2_ 0.999f
#define AEPS_ 1e-8f
#define PI_F 3.14159265358979f

#define AS1 __attribute__((address_space(1)))
#define AS3 __attribute__((address_space(3)))

#if defined(__AMDGCN__) && __has_builtin(__builtin_amdgcn_global_load_async_to_lds_b32) && \
    __has_builtin(__builtin_amdgcn_global_load_async_to_lds_b128)
#define USE_ASYNC 1
#else
#define USE_ASYNC 0
#endif

__device__ __forceinline__ void wait_async_zero() {
#if defined(__AMDGCN__)
#if __has_builtin(__builtin_amdgcn_s_wait_asynccnt)
  __builtin_amdgcn_s_wait_asynccnt(0);
#else
  asm volatile("s_wait_asynccnt 0x0" ::: "memory");
#endif
#endif
}

// ---------------------------------------------------------------------------
// Kernel 0: z = x @ W + bias via V_WMMA_F32_16X16X4_F32.
// Block = 256 threads (8 waves) -> one N-tile (16 cols) x 8 M-tiles (128 rows).
// Double-buffered LDS slabs staged with async global->LDS loads: per K-slab
// we wait + single barrier, kick off the NEXT slab's async loads, then run a
// register-blocked 8-WMMA accumulation chain (all ds_loads hoisted, one wait).
// OOB columns are clamped at staging time (their results are never stored),
// so the hot loop is branch-free with full EXEC (WMMA requirement).
// ---------------------------------------------------------------------------
__global__ __launch_bounds__(256) void aaa_gemm_init(
    const float* __restrict__ x, const float* __restrict__ W,
    const float* __restrict__ bias, float* __restrict__ u,
    float* __restrict__ m, float* __restrict__ v) {
  __shared__ float shW[2][KSLAB * 16];     // [buf][k][n]   2x2 KB
  __shared__ float shX[2][128 * KSLAB];    // [buf][row][k] 2x16 KB

  const int t = threadIdx.x;
  const int wave = t >> 5;
  const int lane = t & 31;
  const int half = lane >> 4;
  const int r = lane & 15;
  const int col0 = blockIdx.x * 16;
  const int rowbase = blockIdx.y * 128;
  const int row0 = rowbase + wave * 16;

  // W staging map: thread t covers shW[b][t] and shW[b][t+256] (conflict-free)
  const int wk = t >> 4;                 // 0..15
  const int wc = t & 15;
  const int colc = (col0 + wc < CC_) ? (col0 + wc) : (CC_ - 1);  // clamp OOB

  // X staging map: 8 threads/row, 4 floats (b128) each, 4 iters of 32 rows
  const int xr = t >> 3;                 // 0..31
  const int xc = (t & 7) * 4;            // 0,4,..,28

  auto stage = [&](int ks, int b) {
#if USE_ASYNC
    __builtin_amdgcn_global_load_async_to_lds_b32(
        (AS1 int*)(W + (size_t)(ks + wk) * CC_ + colc),
        (AS3 int*)&shW[b][wk * 16 + wc], 0, 0);
    __builtin_amdgcn_global_load_async_to_lds_b32(
        (AS1 int*)(W + (size_t)(ks + wk + 16) * CC_ + colc),
        (AS3 int*)&shW[b][(wk + 16) * 16 + wc], 0, 0);
#pragma unroll
    for (int it = 0; it < 4; ++it) {
      const int ri = xr + 32 * it;
      __builtin_amdgcn_global_load_async_to_lds_b128(
          (AS1 v4i_gcc*)(x + (size_t)(rowbase + ri) * DD_ + ks + xc),
          (AS3 v4i_gcc*)&shX[b][ri * KSLAB + xc], 0, 0);
    }
#else
    shW[b][wk * 16 + wc]        = W[(size_t)(ks + wk) * CC_ + colc];
    shW[b][(wk + 16) * 16 + wc] = W[(size_t)(ks + wk + 16) * CC_ + colc];
#pragma unroll
    for (int it = 0; it < 4; ++it) {
      const int ri = xr + 32 * it;
#pragma unroll
      for (int q = 0; q < 4; ++q)
        shX[b][ri * KSLAB + xc + q] = x[(size_t)(rowbase + ri) * DD_ + ks + xc + q];
    }
#endif
  };

  v8f acc = {0.f, 0.f, 0.f, 0.f, 0.f, 0.f, 0.f, 0.f};

  stage(0, 0);  // prologue: slab 0 -> buffer 0

  int buf = 0;
  for (int ks = 0; ks < DD_; ks += KSLAB, buf ^= 1) {
    wait_async_zero();     // this wave's staging of slab (ks) done
    __syncthreads();       // all waves staged slab (ks); prev reads of buf^1 done

    if (ks + KSLAB < DD_) stage(ks + KSLAB, buf ^ 1);  // overlap next slab

    // Hoist all fragments for this slab into registers (batched ds_loads),
    // then run the 8-WMMA accumulation chain back-to-back.
    v2f afrag[8], bfrag[8];
#pragma unroll
    for (int s = 0; s < 8; ++s) {
      const int ka = s * 4 + 2 * half;
      afrag[s].x = shX[buf][(wave * 16 + r) * KSLAB + ka];
      afrag[s].y = shX[buf][(wave * 16 + r) * KSLAB + ka + 1];
      bfrag[s].x = shW[buf][ka * 16 + r];
      bfrag[s].y = shW[buf][(ka + 1) * 16 + r];
    }
#pragma unroll
    for (int s = 0; s < 8; ++s) {
      acc = __builtin_amdgcn_wmma_f32_16x16x4_f32(
          /*neg_a=*/false, afrag[s], /*neg_b=*/false, bfrag[s],
          /*c_mod=*/(short)0, acc, /*reuse_a=*/false, /*reuse_b=*/false);
    }
  }

  const int col = col0 + r;
  if (col < CC_) {
    const float bv = bias[col];
#pragma unroll
    for (int q = 0; q < 8; ++q) {
      const size_t off = (size_t)(row0 + q + 8 * half) * CC_ + col;
      u[off] = acc[q] + bv;
      m[off] = 0.f;
      v[off] = 0.f;
    }
  }
}

// ---------------------------------------------------------------------------
// Kernel 1: per-row init stats from z (== u before any update).
// y = argmax(z); l_org = lse - zmax; l_trg via sin schedule; p_trg = exp(-l_org).
// ---------------------------------------------------------------------------
__global__ void aaa_row_init(const float* __restrict__ u,
                             int* __restrict__ y,
                             float* __restrict__ l_trg,
                             float* __restrict__ p_trg) {
  __shared__ float smax[256];
  __shared__ int   sidx[256];
  __shared__ float ssum[256];
  const int i = blockIdx.x;
  const int t = threadIdx.x;
  const float* row = u + (size_t)i * CC_;

  float mx = -INFINITY; int mi = 0;
  for (int j = t; j < CC_; j += 256) {
    const float vv = row[j];
    if (vv > mx) { mx = vv; mi = j; }
  }
  smax[t] = mx; sidx[t] = mi;
  __syncthreads();
  for (int s = 128; s > 0; s >>= 1) {
    if (t < s) {
      const float o = smax[t + s]; const int oi = sidx[t + s];
      if (o > smax[t] || (o == smax[t] && oi < sidx[t])) { smax[t] = o; sidx[t] = oi; }
    }
    __syncthreads();
  }
  mx = smax[0];

  float se = 0.f;
  for (int j = t; j < CC_; j += 256) se += expf(row[j] - mx);
  ssum[t] = se;
  __syncthreads();
  for (int s = 128; s > 0; s >>= 1) {
    if (t < s) ssum[t] += ssum[t + s];
    __syncthreads();
  }

  if (t == 0) {
    const float lse  = mx + logf(ssum[0]);
    const float lorg = lse - mx;  // CE(z, argmax z)
    const float latr = (floorf(lorg / TAU_) + 0.5f) * TAU_;
    const float lt   = lorg - ALPHA_ * TAU_ *
                       sinf(PI_F * (1.f - 2.f * (lorg - latr) / TAU_));
    y[i]     = sidx[0];
    l_trg[i] = lt;
    p_trg[i] = expf(-lorg);  // max softmax prob of row
  }
}

// ---------------------------------------------------------------------------
// Per-step kernel A: per-row softmax stats of current u.
// ---------------------------------------------------------------------------
__global__ void aaa_row_pass(const float* __restrict__ u,
                             const int* __restrict__ y,
                             float* __restrict__ row_lse,
                             float* __restrict__ row_ce,
                             float* __restrict__ row_maxp,
                             int* __restrict__ row_maxc) {
  __shared__ float smax[256];
  __shared__ int   sidx[256];
  __shared__ float ssum[256];
  const int i = blockIdx.x;
  const int t = threadIdx.x;
  const float* row = u + (size_t)i * CC_;

  float mx = -INFINITY; int mi = 0;
  for (int j = t; j < CC_; j += 256) {
    const float vv = row[j];
    if (vv > mx) { mx = vv; mi = j; }
  }
  smax[t] = mx; sidx[t] = mi;
  __syncthreads();
  for (int s = 128; s > 0; s >>= 1) {
    if (t < s) {
      const float o = smax[t + s]; const int oi = sidx[t + s];
      if (o > smax[t] || (o == smax[t] && oi < sidx[t])) { smax[t] = o; sidx[t] = oi; }
    }
    __syncthreads();
  }
  mx = smax[0];

  float se = 0.f;
  for (int j = t; j < CC_; j += 256) se += expf(row[j] - mx);
  ssum[t] = se;
  __syncthreads();
  for (int s = 128; s > 0; s >>= 1) {
    if (t < s) ssum[t] += ssum[t + s];
    __syncthreads();
  }

  if (t == 0) {
    const float lse = mx + logf(ssum[0]);
    row_lse[i]  = lse;
    row_ce[i]   = lse - row[y[i]];
    row_maxp[i] = expf(mx - lse);   // max softmax prob of this row
    row_maxc[i] = sidx[0];          // its column
  }
}

// ---------------------------------------------------------------------------
// Per-step kernel B (single block): global max prob M, its (r*, c*), and
// S = sum_i sign(M - p_trg_i).
// ---------------------------------------------------------------------------
__global__ void aaa_global_reduce(const float* __restrict__ row_maxp,
                                  const int* __restrict__ row_maxc,
                                  const float* __restrict__ p_trg,
                                  float* __restrict__ gM,
                                  float* __restrict__ gS,
                                  int* __restrict__ grs,
                                  int* __restrict__ gcs) {
  __shared__ float smax[256];
  __shared__ int   sidx[256];
  __shared__ float ssum[256];
  const int t = threadIdx.x;

  float mx = -INFINITY; int mi = 0;
  for (int i = t; i < BB_; i += 256) {
    const float vv = row_maxp[i];
    if (vv > mx) { mx = vv; mi = i; }
  }
  smax[t] = mx; sidx[t] = mi;
  __syncthreads();
  for (int s = 128; s > 0; s >>= 1) {
    if (t < s) {
      const float o = smax[t + s]; const int oi = sidx[t + s];
      if (o > smax[t] || (o == smax[t] && oi < sidx[t])) { smax[t] = o; sidx[t] = oi; }
    }
    __syncthreads();
  }
  const float M = smax[0];

  float ss = 0.f;
  for (int i = t; i < BB_; i += 256) {
    const float d = M - p_trg[i];
    ss += (d > 0.f) ? 1.f : ((d < 0.f) ? -1.f : 0.f);
  }
  ssum[t] = ss;
  __syncthreads();
  for (int s = 128; s > 0; s >>= 1) {
    if (t < s) ssum[t] += ssum[t + s];
    __syncthreads();
  }

  if (t == 0) {
    *gM  = M;
    *gS  = ssum[0];
    *grs = sidx[0];
    *gcs = row_maxc[sidx[0]];
  }
}

// ---------------------------------------------------------------------------
// Per-step kernel C: fused analytic gradient + AdamW update (elementwise).
//   term1: sign(CE_i - l_trg_i) * (p_ij - onehot(y_i))
//   term2 (global-max row r* only): BETA * S * M * (onehot(c*) - p_{r*,j})
// ---------------------------------------------------------------------------
__global__ void aaa_adamw_update(float* __restrict__ u,
                                 float* __restrict__ m,
                                 float* __restrict__ v,
                                 const int* __restrict__ y,
                                 const float* __restrict__ l_trg,
                                 const float* __restrict__ row_lse,
                                 const float* __restrict__ row_ce,
                                 const float* __restrict__ gM,
                                 const float* __restrict__ gS,
                                 const int* __restrict__ grs,
                                 const int* __restrict__ gcs,
                                 float bc1inv, float bc2inv) {
  const int i = blockIdx.y;
  const int j = blockIdx.x * blockDim.x + threadIdx.x;
  if (j >= CC_) return;
  const size_t off = (size_t)i * CC_ + j;

  const float ui = u[off];
  const float p  = expf(ui - row_lse[i]);
  const float d1 = row_ce[i] - l_trg[i];
  const float si = (d1 > 0.f) ? 1.f : ((d1 < 0.f) ? -1.f : 0.f);
  float g = si * (p - ((j == y[i]) ? 1.f : 0.f));
  if (i == *grs) {
    const float M = *gM;
    g += BETA_ * (*gS) * M * (((j == *gcs) ? 1.f : 0.f) - p);
  }

  const float un = ui * (1.f - LR_ * WD_);   // decoupled weight decay
  const float mn = AB1_ * m[off] + (1.f - AB1_) * g;
  const float vn = AB2_ * v[off] + (1.f - AB2_) * g * g;
  m[off] = mn;
  v[off] = vn;
  u[off] = un - LR_ * (mn * bc1inv) / (sqrtf(vn * bc2inv) + AEPS_);
}

// ---------------------------------------------------------------------------
extern "C" void kernel_launch(void* const* d_in, const int* in_sizes, int n_in,
                              void* d_out, int out_size, void* d_ws, size_t ws_size,
                              hipStream_t stream) {
  (void)in_sizes; (void)n_in; (void)out_size; (void)ws_size;

  const float* x    = (const float*)d_in[0];
  const float* W    = (const float*)d_in[1];
  const float* bias = (const float*)d_in[2];
  float* u = (float*)d_out;  // protected logits live directly in the output

  const size_t NEL = (size_t)BB_ * CC_;
  float* ws_f      = (float*)d_ws;
  float* m         = ws_f;            // [B*C]
  float* v         = m + NEL;         // [B*C]
  float* l_trg     = v + NEL;         // [B]
  float* p_trg     = l_trg + BB_;     // [B]
  float* row_lse   = p_trg + BB_;     // [B]
  float* row_ce    = row_lse + BB_;   // [B]
  float* row_maxp  = row_ce + BB_;    // [B]
  float* gM        = row_maxp + BB_;  // [1]
  float* gS        = gM + 1;          // [1]
  int*   y         = (int*)(gS + 1);  // [B]
  int*   row_maxc  = y + BB_;         // [B]
  int*   grs       = row_maxc + BB_;  // [1]
  int*   gcs       = grs + 1;         // [1]

  // GEMM: grid = (63 N-tiles, 32 M-chunks of 128 rows), 256 threads / block
  aaa_gemm_init<<<dim3(NTILE, 32), 256, 0, stream>>>(x, W, bias, u, m, v);
  aaa_row_init<<<BB_, 256, 0, stream>>>(u, y, l_trg, p_trg);

  double b1t = 1.0, b2t = 1.0;
  for (int t = 1; t <= 100; ++t) {
    b1t *= (double)AB1_;
    b2t *= (double)AB2_;
    const float bc1inv = (float)(1.0 / (1.0 - b1t));
    const float bc2inv = (float)(1.0 / (1.0 - b2t));
    aaa_row_pass<<<BB_, 256, 0, stream>>>(u, y, row_lse, row_ce, row_maxp, row_maxc);
    aaa_global_reduce<<<1, 256, 0, stream>>>(row_maxp, row_maxc, p_trg, gM, gS, grs, gcs);
    aaa_adamw_update<<<dim3(4, BB_), 256, 0, stream>>>(
        u, m, v, y, l_trg, row_lse, row_ce, gM, gS, grs, gcs, bc1inv, bc2inv);
  }
}